// BlockLinear_63428077027579
// MI455X (gfx1250) — compile-verified
//
#include <hip/hip_runtime.h>
#include <math.h>

// BlockLinear: y[b,o] = sum_{k<8} exp(lw[o,k]) * x[b, 8o+k]
// B=16384, IN=8192, O=1024, K=8.  ~576MB moved, ~0.27GFLOP -> pure HBM
// bandwidth problem (~25us at 23.3TB/s).  WMMA is structurally inapplicable
// (block-diagonal weight: no shared K axis across N columns without 16x
// zero-padding), so the CDNA5 feature used is the async global->LDS
// copy pipeline (ASYNCcnt) for prefetch at zero VGPR-return cost.

#define IN_FEAT 8192
#define O_FEAT  1024
#define BSZ     8
#define RPT     16      // rows of B handled per thread
#define TPB     256

typedef float f4 __attribute__((ext_vector_type(4)));
// Builtin parameter types (from the round-1 diagnostic): pointer to
// GCC-vector int4, in __device__ (AS1) / __shared__ (AS3) address spaces.
typedef int v4i __attribute__((vector_size(16)));
typedef __attribute__((address_space(1))) v4i* as1_v4i;
typedef __attribute__((address_space(3))) v4i* as3_v4i;

#if defined(__HIP_DEVICE_COMPILE__) && defined(__gfx1250__)
  #define WAIT_ASYNC(n) asm volatile("s_wait_asynccnt %0" :: "i"(n) : "memory")
  #if __has_builtin(__builtin_amdgcn_global_load_async_to_lds_b128)
    #define ASYNC_B128(gp, lp)                                                  \
      __builtin_amdgcn_global_load_async_to_lds_b128(                           \
          (as1_v4i)(uintptr_t)(gp),                                             \
          (as3_v4i)(unsigned)(uintptr_t)(lp), 0, 0)
  #else
    // ISA mnemonic fallback: VDST = VGPR holding LDS byte address,
    // VADDR = 64-bit global address VGPR pair (GV mode).
    #define ASYNC_B128(gp, lp)                                                  \
      asm volatile("global_load_async_to_lds_b128 %0, %1, off"                  \
                   :: "v"((unsigned)(uintptr_t)(lp)),                           \
                      "v"((unsigned long long)(uintptr_t)(gp))                  \
                   : "memory")
  #endif
  #define USE_ASYNC 1
#else
  #define USE_ASYNC 0
#endif

__global__ __launch_bounds__(TPB) void block_linear_kernel(
    const float* __restrict__ x, const float* __restrict__ logw,
    float* __restrict__ out)
{
    const int t = threadIdx.x;
    const int g = blockIdx.x * TPB + t;
    const int o = g & (O_FEAT - 1);
    const long bbase = (long)(g >> 10) * RPT;

    // Per-output weights: 8 floats, exp'd once, held in VGPRs for 16 rows.
    const f4 lw0 = *(const f4*)(logw + o * BSZ);
    const f4 lw1 = *(const f4*)(logw + o * BSZ + 4);
    f4 w0, w1;
    #pragma unroll
    for (int j = 0; j < 4; ++j) { w0[j] = expf(lw0[j]); w1[j] = expf(lw1[j]); }

    const float* xrow = x   + bbase * IN_FEAT + (long)o * BSZ;  // 32B aligned
    float*       orow = out + bbase * O_FEAT  + o;

#if USE_ASYNC
    // [buffer(2)][chunk(2)][thread(TPB)] 16B slots: lane stride 16B ->
    // conflict-minimal ds_load_b128.  16KB LDS per block.
    __shared__ f4 lds[4 * TPB];
    f4* const s00 = &lds[0 * TPB + t];
    f4* const s01 = &lds[1 * TPB + t];
    f4* const s10 = &lds[2 * TPB + t];
    f4* const s11 = &lds[3 * TPB + t];

    // prologue: fill buffer 0 with row 0
    ASYNC_B128(xrow,     s00);
    ASYNC_B128(xrow + 4, s01);

    #pragma unroll
    for (int i = 0; i < RPT; ++i) {
        const int p = i & 1;
        if (i + 1 < RPT) {
            const float* nxt = xrow + (long)(i + 1) * IN_FEAT;
            if (p == 0) { ASYNC_B128(nxt, s10); ASYNC_B128(nxt + 4, s11); }
            else        { ASYNC_B128(nxt, s00); ASYNC_B128(nxt + 4, s01); }
            WAIT_ASYNC(2);      // 2 just-issued stay in flight; current buf done
        } else {
            WAIT_ASYNC(0);
        }
        const f4 a = p ? *s10 : *s00;
        const f4 b = p ? *s11 : *s01;

        float acc = a[0] * w0[0];
        acc = fmaf(a[1], w0[1], acc);
        acc = fmaf(a[2], w0[2], acc);
        acc = fmaf(a[3], w0[3], acc);
        acc = fmaf(b[0], w1[0], acc);
        acc = fmaf(b[1], w1[1], acc);
        acc = fmaf(b[2], w1[2], acc);
        acc = fmaf(b[3], w1[3], acc);
        orow[(long)i * O_FEAT] = acc;
    }
#else
    // Portable fallback (host pass / non-gfx1250): direct b128 streaming.
    #pragma unroll
    for (int i = 0; i < RPT; ++i) {
        const f4 a = *(const f4*)(xrow + (long)i * IN_FEAT);
        const f4 b = *(const f4*)(xrow + (long)i * IN_FEAT + 4);
        float acc = a[0] * w0[0];
        acc = fmaf(a[1], w0[1], acc);
        acc = fmaf(a[2], w0[2], acc);
        acc = fmaf(a[3], w0[3], acc);
        acc = fmaf(b[0], w1[0], acc);
        acc = fmaf(b[1], w1[1], acc);
        acc = fmaf(b[2], w1[2], acc);
        acc = fmaf(b[3], w1[3], acc);
        orow[(long)i * O_FEAT] = acc;
    }
#endif
}

extern "C" void kernel_launch(void* const* d_in, const int* in_sizes, int n_in,
                              void* d_out, int out_size, void* d_ws, size_t ws_size,
                              hipStream_t stream) {
    (void)n_in; (void)out_size; (void)d_ws; (void)ws_size;
    const float* x    = (const float*)d_in[0];   // [B, 8192] f32
    const float* logw = (const float*)d_in[1];   // [1024, 8] f32
    float*       out  = (float*)d_out;           // [B, 1024] f32

    const int B = in_sizes[0] / IN_FEAT;                 // 16384
    const int total_threads = O_FEAT * (B / RPT);        // 1,048,576
    const int blocks = total_threads / TPB;              // 4096

    block_linear_kernel<<<blocks, TPB, 0, stream>>>(x, logw, out);
}